// MultiVectorFieldModel_50603304682175
// MI455X (gfx1250) — compile-verified
//
#include <hip/hip_runtime.h>
#include <math.h>

// ---------------------------------------------------------------------------
// Problem constants (from the reference)
// ---------------------------------------------------------------------------
#define N_ROWS    1000000
#define DATA_DIM  16
#define IN_DIM    17          // DATA_DIM + 1 (t)
#define HIDDEN    6
#define N_BUCKETS 8
#define N_MODELS  65          // N_COND*N_BUCKETS + 1
#define X_COLS    18          // [data(16), cond, t]

#define PADN      (N_ROWS + N_MODELS * 16)   // perm array length (tile padded)
#define MAX_TILES (PADN / 16)
#define WAVES_PER_BLOCK 8

// ---------------- workspace layout ----------------
// int region
#define OFF_COUNTS 0                       // [65]
#define OFF_FILL   65                      // [65]
#define OFF_POFF   130                     // [66] padded exclusive prefix
#define OFF_TOTAL  196                     // [1]  total tiles
#define OFF_MIDS   256                     // [N_ROWS]
#define OFF_PERM   (256 + N_ROWS)          // [PADN]
#define WS_INTS    (OFF_PERM + PADN)       // (even -> float region stays 8B aligned)
// float region (starts at ws + WS_INTS ints)
#define PK1_OFF   0                        // [65][5][32][2]  W1 in B-lane layout
#define PK1_SZ    (N_MODELS * 5 * 64)
#define PK2_OFF   (PK1_OFF + PK1_SZ)       // [65][2][32][2]  W2 in B-lane layout
#define PK2_SZ    (N_MODELS * 2 * 64)
#define PB1_OFF   (PK2_OFF + PK2_SZ)       // [65][32]        b1 padded per lane
#define PB1_SZ    (N_MODELS * 32)
#define SINK_OFF  (PB1_OFF + PB1_SZ)       // [64] garbage sink for padded rows
#define FP_FLOATS (SINK_OFF + 64)
#define PACK_ELEMS (PK1_SZ + PK2_SZ + PB1_SZ)

typedef float v2f __attribute__((ext_vector_type(2)));
typedef float v8f __attribute__((ext_vector_type(8)));

__device__ __forceinline__ int route(float condf, float t) {
  int c = (int)condf;
  int b = (int)floorf(t * (float)N_BUCKETS);
  b = b < 0 ? 0 : (b > N_BUCKETS - 1 ? N_BUCKETS - 1 : b);
  return (c == 0) ? 0 : 1 + (c - 1) * N_BUCKETS + b;
}

// ---------------------------------------------------------------------------
// Pass 0a: init counters and perm sentinel
// ---------------------------------------------------------------------------
__global__ void init_kernel(int* __restrict__ ws) {
  int i = blockIdx.x * blockDim.x + threadIdx.x;
  if (i < PADN) ws[OFF_PERM + i] = -1;
  if (i < N_MODELS) { ws[OFF_COUNTS + i] = 0; ws[OFF_FILL + i] = 0; }
  if (i == 0) ws[OFF_TOTAL] = 0;
}

// ---------------------------------------------------------------------------
// Pass 0b: prepack weights into the exact per-lane WMMA B-fragment layout,
// zero padding baked in.  B layout: col n = lane%16, K = reg + 2*(lane>=16).
//   PK1[mid][s][lane][j]: k = 4s + 2*(lane>>4) + j  (K padded 17->20, N 6->16)
//   PK2[mid][s][lane][j]: k = 4s + 2*(lane>>4) + j  (K padded 6->8)
//   PB1[mid][lane]      : b1[mid][n] for n<6 else 0
// ---------------------------------------------------------------------------
__global__ void pack_kernel(const float* __restrict__ W1, const float* __restrict__ W2,
                            const float* __restrict__ b1, float* __restrict__ fp) {
  int i = blockIdx.x * blockDim.x + threadIdx.x;
  if (i >= PACK_ELEMS) return;
  if (i < PK1_SZ) {
    int mid = i / 320, rem = i % 320;
    int s = rem / 64, q = rem % 64, lane = q >> 1, j = q & 1;
    int k = 4 * s + 2 * (lane >> 4) + j, n = lane & 15;
    fp[PK1_OFF + i] = (k < IN_DIM && n < HIDDEN) ? W1[mid * (IN_DIM * HIDDEN) + k * HIDDEN + n] : 0.0f;
  } else if (i < PK1_SZ + PK2_SZ) {
    int i2 = i - PK1_SZ;
    int mid = i2 / 128, rem = i2 % 128;
    int s = rem / 64, q = rem % 64, lane = q >> 1, j = q & 1;
    int k = 4 * s + 2 * (lane >> 4) + j, n = lane & 15;
    fp[PK2_OFF + i2] = (k < HIDDEN) ? W2[mid * (HIDDEN * DATA_DIM) + k * DATA_DIM + n] : 0.0f;
  } else {
    int i3 = i - PK1_SZ - PK2_SZ;
    int mid = i3 / 32, lane = i3 % 32, n = lane & 15;
    fp[PB1_OFF + i3] = (n < HIDDEN) ? b1[mid * HIDDEN + n] : 0.0f;
  }
}

// ---------------------------------------------------------------------------
// Pass 1: route each row, histogram per model
// ---------------------------------------------------------------------------
__global__ void route_count_kernel(const float* __restrict__ x, int* __restrict__ ws) {
  int i = blockIdx.x * blockDim.x + threadIdx.x;
  if (i >= N_ROWS) return;
  float condf = x[(size_t)i * X_COLS + 16];
  float t     = x[(size_t)i * X_COLS + 17];
  int mid = route(condf, t);
  ws[OFF_MIDS + i] = mid;
  atomicAdd(&ws[OFF_COUNTS + mid], 1);
}

// ---------------------------------------------------------------------------
// Pass 2: tiny serial prefix sum (65 entries), padded to 16-row tiles
// ---------------------------------------------------------------------------
__global__ void scan_kernel(int* __restrict__ ws) {
  if (threadIdx.x == 0 && blockIdx.x == 0) {
    int run = 0;
    for (int m = 0; m < N_MODELS; ++m) {
      ws[OFF_POFF + m] = run;
      run += ((ws[OFF_COUNTS + m] + 15) >> 4) << 4;
    }
    ws[OFF_POFF + N_MODELS] = run;
    ws[OFF_TOTAL] = run >> 4;
  }
}

// ---------------------------------------------------------------------------
// Pass 3: scatter row indices into per-model segments
// ---------------------------------------------------------------------------
__global__ void scatter_kernel(int* __restrict__ ws) {
  int i = blockIdx.x * blockDim.x + threadIdx.x;
  if (i >= N_ROWS) return;
  int mid = ws[OFF_MIDS + i];
  int pos = atomicAdd(&ws[OFF_FILL + mid], 1);
  ws[OFF_PERM + ws[OFF_POFF + mid] + pos] = i;
}

// ---------------------------------------------------------------------------
// Pass 4: one wave32 per 16-row tile; fp32 WMMA for both layers.
//   GEMM1: (16 x 20pad) x (20pad x 16) via 5x v_wmma_f32_16x16x4_f32
//   GEMM2: (16 x 8pad)  x (8pad  x 16) via 2x v_wmma_f32_16x16x4_f32
// A layout:  row = lane%16,            K = reg + 2*(lane>=16)
// B layout:  col = lane%16,            K = reg + 2*(lane>=16)
// C layout:  M = reg + 8*(lane>=16),   N = lane%16
// All operand fetches are unconditional b64 loads + cndmask (no exec churn).
// ---------------------------------------------------------------------------
__global__ __launch_bounds__(32 * WAVES_PER_BLOCK)
void mlp_wmma_kernel(const float* __restrict__ x,
                     const float* __restrict__ b2, float* __restrict__ out,
                     const int* __restrict__ ws, float* __restrict__ fp) {
  __shared__ float hbuf[WAVES_PER_BLOCK][16][16];

  const int lane = threadIdx.x & 31;
  const int wave = threadIdx.x >> 5;
  const int tile = blockIdx.x * WAVES_PER_BLOCK + wave;
  const int total = ws[OFF_TOTAL];
  const int* poff = ws + OFF_POFF;
  const int* perm = ws + OFF_PERM;
  const bool active = (tile < total);   // wave-uniform -> EXEC all-1s inside

  const int row0  = tile * 16;
  const int m     = lane & 15;          // A row / C col handled by this lane
  const int khalf = (lane >> 4) * 2;    // K sub-offset from lane half

  int mid = 0;

  if (active) {
    // model owning this tile: largest mid with poff[mid] <= row0 (wave-uniform)
    int lo = 0, hi = N_MODELS - 1;
    while (lo < hi) {
      int md = (lo + hi + 1) >> 1;
      if (poff[md] <= row0) lo = md; else hi = md - 1;
    }
    mid = __builtin_amdgcn_readfirstlane(lo);

    const int pr  = perm[row0 + m];     // source row for A row m (-1 = pad)
    const int prc = pr >= 0 ? pr : 0;   // clamped: loads always legal
    const float* xr = x + (size_t)prc * X_COLS;
    const v2f zero2 = {0.0f, 0.0f};

    // ---- gather all layer-1 fragments (10 unconditional b64 loads) ----
    const v2f* B1p = (const v2f*)(fp + PK1_OFF) + (size_t)mid * 160 + lane;
    v2f a[5], b[5];
#pragma unroll
    for (int s = 0; s < 5; ++s) b[s] = B1p[s * 32];
#pragma unroll
    for (int s = 0; s < 4; ++s) {
      v2f av = *(const v2f*)(xr + 4 * s + khalf);   // 8B aligned column pair
      a[s] = (pr >= 0) ? av : zero2;
    }
    {
      v2f ct = *(const v2f*)(xr + 16);              // cols {cond, t}
      float tval = ((lane < 16) && (pr >= 0)) ? ct.y : 0.0f;  // k=16 -> t
      a[4] = (v2f){tval, 0.0f};                     // k>=17 pad
    }

    v8f c = {};
#pragma unroll
    for (int s = 0; s < 5; ++s)
      c = __builtin_amdgcn_wmma_f32_16x16x4_f32(false, a[s], false, b[s],
                                                (short)0, c, false, false);

    // bias + ReLU, park h tile in LDS to re-layout C -> A for layer 2
    const float bias = fp[PB1_OFF + mid * 32 + lane];   // prepadded (0 for n>=6)
#pragma unroll
    for (int r = 0; r < 8; ++r) {
      const int M = r + 8 * (lane >> 4);
      hbuf[wave][M][m] = fmaxf(c[r] + bias, 0.0f);      // cols >= HIDDEN stay 0
    }
  }

  __syncthreads();   // all waves reach this (inactive waves skip work only)

  if (active) {
    const v2f* B2p = (const v2f*)(fp + PK2_OFF) + (size_t)mid * 64 + lane;
    v2f a2[2], bb[2];
#pragma unroll
    for (int s = 0; s < 2; ++s) {
      bb[s] = B2p[s * 32];
      a2[s] = *(const v2f*)(&hbuf[wave][m][4 * s + khalf]);   // ds_load_b64
    }
    v8f c2 = {};
#pragma unroll
    for (int s = 0; s < 2; ++s)
      c2 = __builtin_amdgcn_wmma_f32_16x16x4_f32(false, a2[s], false, bb[s],
                                                 (short)0, c2, false, false);

    const float bo = b2[(size_t)mid * DATA_DIM + m];
    float* snk = fp + SINK_OFF + lane;
#pragma unroll
    for (int r = 0; r < 8; ++r) {
      const int M   = r + 8 * (lane >> 4);
      const int pr2 = perm[row0 + M];
      const int pc2 = pr2 >= 0 ? pr2 : 0;
      float* dst = (pr2 >= 0) ? (out + (size_t)pc2 * DATA_DIM + m) : snk;
      *dst = c2[r] + bo;                              // branchless scatter
    }
  }
}

// ---------------------------------------------------------------------------
// Fallback: fused scalar kernel (used only if workspace is too small).
// ---------------------------------------------------------------------------
__global__ void mlp_scalar_kernel(const float* __restrict__ x,
                                  const float* __restrict__ W1, const float* __restrict__ b1,
                                  const float* __restrict__ W2, const float* __restrict__ b2,
                                  float* __restrict__ out) {
  int i = blockIdx.x * blockDim.x + threadIdx.x;
  if (i >= N_ROWS) return;
  const float* xr = x + (size_t)i * X_COLS;
  float xin[IN_DIM];
#pragma unroll
  for (int j = 0; j < DATA_DIM; ++j) xin[j] = xr[j];
  const float condf = xr[16];
  const float t     = xr[17];
  xin[16] = t;
  const int mid = route(condf, t);
  const float* W1m = W1 + mid * IN_DIM * HIDDEN;
  const float* b1m = b1 + mid * HIDDEN;
  const float* W2m = W2 + mid * HIDDEN * DATA_DIM;
  const float* b2m = b2 + mid * DATA_DIM;
  float h[HIDDEN];
#pragma unroll
  for (int hj = 0; hj < HIDDEN; ++hj) {
    float acc = b1m[hj];
#pragma unroll
    for (int k = 0; k < IN_DIM; ++k) acc = fmaf(xin[k], W1m[k * HIDDEN + hj], acc);
    h[hj] = fmaxf(acc, 0.0f);
  }
#pragma unroll
  for (int d = 0; d < DATA_DIM; ++d) {
    float acc = b2m[d];
#pragma unroll
    for (int k = 0; k < HIDDEN; ++k) acc = fmaf(h[k], W2m[k * DATA_DIM + d], acc);
    out[(size_t)i * DATA_DIM + d] = acc;
  }
}

// ---------------------------------------------------------------------------
extern "C" void kernel_launch(void* const* d_in, const int* in_sizes, int n_in,
                              void* d_out, int out_size, void* d_ws, size_t ws_size,
                              hipStream_t stream) {
  const float* x  = (const float*)d_in[0];
  const float* W1 = (const float*)d_in[1];
  const float* b1 = (const float*)d_in[2];
  const float* W2 = (const float*)d_in[3];
  const float* b2 = (const float*)d_in[4];
  float* out = (float*)d_out;

  const size_t need = (size_t)WS_INTS * sizeof(int) + (size_t)FP_FLOATS * sizeof(float);
  if (ws_size < need) {
    mlp_scalar_kernel<<<(N_ROWS + 255) / 256, 256, 0, stream>>>(x, W1, b1, W2, b2, out);
    return;
  }

  int* ws = (int*)d_ws;
  float* fp = (float*)(ws + WS_INTS);

  init_kernel<<<(PADN + 255) / 256, 256, 0, stream>>>(ws);
  pack_kernel<<<(PACK_ELEMS + 255) / 256, 256, 0, stream>>>(W1, W2, b1, fp);
  route_count_kernel<<<(N_ROWS + 255) / 256, 256, 0, stream>>>(x, ws);
  scan_kernel<<<1, 32, 0, stream>>>(ws);
  scatter_kernel<<<(N_ROWS + 255) / 256, 256, 0, stream>>>(ws);
  mlp_wmma_kernel<<<(MAX_TILES + WAVES_PER_BLOCK - 1) / WAVES_PER_BLOCK,
                    32 * WAVES_PER_BLOCK, 0, stream>>>(x, b2, out, ws, fp);
}